// Sampler_37752762532393
// MI455X (gfx1250) — compile-verified
//
#include <hip/hip_runtime.h>
#include <hip/hip_bf16.h>
#include <math.h>

typedef __attribute__((ext_vector_type(2)))  float        v2f;
typedef __attribute__((ext_vector_type(8)))  float        v8f;
typedef __attribute__((ext_vector_type(4)))  unsigned int v4u;
typedef __attribute__((ext_vector_type(4)))  int          v4i;
typedef __attribute__((ext_vector_type(8)))  int          v8i;

#define N_ROWS 128
#define VOCAB  50257
#define HID    4096
#define LSEQ   200
#define NB     ((VOCAB + 63) / 64)   // 786 column stripes of 64
#define VPAD   (NB * 64)             // 50304
#define CAND   2048

#define KC      32                   // K-chunk staged per TDM transfer
#define NCHUNK  (HID / KC)           // 128
#define ROWDW   (KC + 2)             // 34 dwords/row in LDS (TDM pad: conflict-free)

// Monotone float->uint key: ascending key order == ascending float order.
__device__ __forceinline__ unsigned fkey(float f) {
    unsigned b = __float_as_uint(f);
    return (b & 0x80000000u) ? ~b : (b | 0x80000000u);
}

// Low 32 bits of a generic pointer into the LDS aperture == LDS byte offset
// (see ISA 10.2).
__device__ __forceinline__ unsigned lds_off(const void* p) {
    return (unsigned)(size_t)p;
}

// ---------------------------------------------------------------------------
// TDM issue: 2D tile (tile_d1 rows x tile_d0 elems, fp32) from global to LDS.
// D# per ISA ch.8: group0 = {count=1, lds_addr, global_addr, type=2},
// group1 packs data_size=4B, pad_enable (interval 32 dwords, amount 2 dwords),
// tensor dims/tile dims, 48-bit dim0 stride. Groups 2/3 zero (2D tensor).
// This toolchain's builtin takes 6 args (g0, g1, g2, g3, extra group, cpol).
// ---------------------------------------------------------------------------
__device__ __forceinline__ void tdm_load_2d(unsigned lds_addr, const void* gptr,
                                            int tile_d0, int tile_d1,
                                            long long stride0_elems)
{
    unsigned long long ga = (unsigned long long)(size_t)gptr;
    v4u g0;
    g0[0] = 1u;                                        // count=1, user descriptor
    g0[1] = lds_addr;                                  // lds_addr[31:0]
    g0[2] = (unsigned)(ga & 0xffffffffull);            // global_addr[31:0]
    g0[3] = (unsigned)((ga >> 32) & 0x01ffffffull)     // global_addr[56:32]
          | 0x80000000u;                               // type=2 ("image")
    v8i g1;
    g1[0] = (2 << 16)      // data_size = 4 bytes
          | (1 << 20)      // pad_enable
          | (4 << 22)      // pad_interval: 32 dwords
          | (1 << 25);     // pad_amount: 2 dwords
    g1[1] = (tile_d0 & 0xffff) << 16;                          // tensor_dim0 lo16
    g1[2] = ((tile_d0 >> 16) & 0xffff)                         // tensor_dim0 hi
          | ((tile_d1 & 0xffff) << 16);                        // tensor_dim1 lo16
    g1[3] = ((tile_d1 >> 16) & 0xffff)                         // tensor_dim1 hi
          | ((tile_d0 & 0xffff) << 16);                        // tile_dim0
    g1[4] = (tile_d1 & 0xffff);                                // tile_dim1 (tile_dim2=0)
    g1[5] = (int)(stride0_elems & 0xffffffffll);               // dim0 stride lo32
    g1[6] = (int)((stride0_elems >> 32) & 0xffffll);           // dim0 stride hi16
    g1[7] = 0;
    v4i gz4 = {0, 0, 0, 0};
    v8i gz8 = {0, 0, 0, 0, 0, 0, 0, 0};
    __builtin_amdgcn_tensor_load_to_lds(g0, g1, gz4, gz4, gz8, 0);
}

// ---------------------------------------------------------------------------
// Kernel 1: logits = hidden @ emb^T via V_WMMA_F32_16X16X4_F32.
// Block = 256 threads (8 wave32), computes a 128x64 stripe of logits.
// TDM double-buffers A(128xKC) and B(64xKC) chunks into LDS; each wave owns a
// 2x2 grid of 16x16 tiles -> 1 ds_load_b64 per WMMA.
// LDS rows are padded to 34 dwords by the TDM (34*i mod 64 injective for
// i<16 -> fragment loads are bank-conflict-free).
// ---------------------------------------------------------------------------
__global__ __launch_bounds__(256)
void sampler_gemm_kernel(const float* __restrict__ hs,
                         const float* __restrict__ emb,
                         float* __restrict__ logits)
{
    __shared__ float abuf[2][N_ROWS * ROWDW];   // 2 x 17408 B
    __shared__ float bbuf[2][64 * ROWDW];       // 2 x  8704 B

    const int wave  = threadIdx.x >> 5;
    const int lane  = threadIdx.x & 31;
    const int lhalf = lane >> 4;                 // 0 or 1
    const int l15   = lane & 15;
    const int p     = wave & 3;                  // m-pair: rows 32p..32p+31
    const int q     = wave >> 2;                 // n-pair: cols 32q..32q+31

    const int v0    = blockIdx.x * 64;
    const int nrows = (VOCAB - v0 < 64) ? (VOCAB - v0) : 64;

    v8f acc[2][2];
    #pragma unroll
    for (int i = 0; i < 2; ++i)
        #pragma unroll
        for (int j = 0; j < 2; ++j)
            acc[i][j] = (v8f){0.f,0.f,0.f,0.f,0.f,0.f,0.f,0.f};

    // Prologue: stage chunk 0.
    if (wave == 0) {
        tdm_load_2d(lds_off(&abuf[0][0]), hs,                          KC, N_ROWS, HID);
        tdm_load_2d(lds_off(&bbuf[0][0]), emb + (size_t)v0 * HID,      KC, nrows,  HID);
        __builtin_amdgcn_s_wait_tensorcnt(0);
    }
    __syncthreads();

    for (int c = 0; c < NCHUNK; ++c) {
        const int buf = c & 1;
        if (wave == 0 && c + 1 < NCHUNK) {
            const int k0 = (c + 1) * KC;
            tdm_load_2d(lds_off(&abuf[buf ^ 1][0]), hs + k0,                     KC, N_ROWS, HID);
            tdm_load_2d(lds_off(&bbuf[buf ^ 1][0]), emb + (size_t)v0 * HID + k0, KC, nrows,  HID);
        }

        const float* __restrict__ A = abuf[buf];
        const float* __restrict__ B = bbuf[buf];
        const int arow0 = (p * 32 + l15) * ROWDW;        // m-tile 2p
        const int arow1 = (p * 32 + 16 + l15) * ROWDW;   // m-tile 2p+1
        const int brow0 = (q * 32 + l15) * ROWDW;        // n-tile 2q
        const int brow1 = (q * 32 + 16 + l15) * ROWDW;   // n-tile 2q+1

        #pragma unroll
        for (int kk = 0; kk < KC; kk += 4) {
            const int ko = kk + (lhalf << 1);            // even -> 8B aligned
            v2f a0 = *(const v2f*)(A + arow0 + ko);
            v2f a1 = *(const v2f*)(A + arow1 + ko);
            v2f b0 = *(const v2f*)(B + brow0 + ko);
            v2f b1 = *(const v2f*)(B + brow1 + ko);
            acc[0][0] = __builtin_amdgcn_wmma_f32_16x16x4_f32(false, a0, false, b0, (short)0, acc[0][0], false, false);
            acc[0][1] = __builtin_amdgcn_wmma_f32_16x16x4_f32(false, a0, false, b1, (short)0, acc[0][1], false, false);
            acc[1][0] = __builtin_amdgcn_wmma_f32_16x16x4_f32(false, a1, false, b0, (short)0, acc[1][0], false, false);
            acc[1][1] = __builtin_amdgcn_wmma_f32_16x16x4_f32(false, a1, false, b1, (short)0, acc[1][1], false, false);
        }

        if (wave == 0 && c + 1 < NCHUNK)
            __builtin_amdgcn_s_wait_tensorcnt(0);
        __syncthreads();
    }

    // Store: C/D layout -> VGPR r holds (M = r + 8*lhalf, N = l15).
    #pragma unroll
    for (int mi = 0; mi < 2; ++mi) {
        #pragma unroll
        for (int ni = 0; ni < 2; ++ni) {
            const int col = v0 + q * 32 + ni * 16 + l15;
            if (col < VOCAB) {
                float* outp = logits + (size_t)(p * 32 + mi * 16 + lhalf * 8) * VPAD + col;
                #pragma unroll
                for (int r = 0; r < 8; ++r)
                    outp[(size_t)r * VPAD] = acc[mi][ni][r];
            }
        }
    }
}

// ---------------------------------------------------------------------------
// Kernel 2: presence/frequency/repetition penalties. One block per row.
// ---------------------------------------------------------------------------
__global__ __launch_bounds__(256)
void sampler_penalty_kernel(const int* __restrict__ toks,
                            const float* __restrict__ pres,
                            const float* __restrict__ freq,
                            const float* __restrict__ rep,
                            float* __restrict__ logits)
{
    const int n = blockIdx.x;
    __shared__ int tok[LSEQ];
    for (int t = threadIdx.x; t < LSEQ; t += blockDim.x)
        tok[t] = toks[n * LSEQ + t];
    __syncthreads();

    for (int t = threadIdx.x; t < LSEQ; t += blockDim.x) {
        const int mytok = tok[t];
        int cnt = 0, first = LSEQ;
        for (int j = 0; j < LSEQ; ++j) {
            if (tok[j] == mytok) { cnt++; if (j < first) first = j; }
        }
        if (first == t) {
            float* lp = logits + (size_t)n * VPAD + mytok;
            float l = *lp;
            const float r = rep[n];
            l = (l > 0.f) ? (l / r) : (l * r);
            l -= freq[n] * (float)cnt;
            l -= pres[n];
            *lp = l;
        }
    }
}

// ---------------------------------------------------------------------------
// Kernel 3: per-row max and full softmax denominator of val = logit/T.
// ---------------------------------------------------------------------------
__global__ __launch_bounds__(256)
void sampler_stats_kernel(const float* __restrict__ logits,
                          const float* __restrict__ temps,
                          float* __restrict__ rowMax,
                          float* __restrict__ rowSum)
{
    const int n = blockIdx.x;
    const float invT = 1.0f / temps[n];
    const float* __restrict__ lp = logits + (size_t)n * VPAD;
    __shared__ float red[256];

    float mx = -INFINITY;
    for (int v = threadIdx.x; v < VOCAB; v += 256)
        mx = fmaxf(mx, lp[v] * invT);
    red[threadIdx.x] = mx;
    __syncthreads();
    for (int s = 128; s > 0; s >>= 1) {
        if (threadIdx.x < s) red[threadIdx.x] = fmaxf(red[threadIdx.x], red[threadIdx.x + s]);
        __syncthreads();
    }
    mx = red[0];
    __syncthreads();

    float sum = 0.f;
    for (int v = threadIdx.x; v < VOCAB; v += 256)
        sum += expf(lp[v] * invT - mx);
    red[threadIdx.x] = sum;
    __syncthreads();
    for (int s = 128; s > 0; s >>= 1) {
        if (threadIdx.x < s) red[threadIdx.x] += red[threadIdx.x + s];
        __syncthreads();
    }
    if (threadIdx.x == 0) { rowMax[n] = mx; rowSum[n] = red[0]; }
}

// ---------------------------------------------------------------------------
// Kernel 4: exact top-k/top-p cutoff per row (radix-select + bitonic sort).
// ---------------------------------------------------------------------------
__global__ __launch_bounds__(256)
void sampler_select_kernel(const float* __restrict__ logits,
                           const float* __restrict__ temps,
                           const int*   __restrict__ topks,
                           const float* __restrict__ topps,
                           const float* __restrict__ rowMax,
                           const float* __restrict__ rowSum,
                           float* __restrict__ rowT,
                           int*   __restrict__ rowVt,
                           float* __restrict__ rowSkept)
{
    const int n = blockIdx.x;
    const float invT = 1.0f / temps[n];
    const float* __restrict__ lp = logits + (size_t)n * VPAD;

    __shared__ int      hist[256];
    __shared__ unsigned sh_prefix;
    __shared__ int      sh_hicount;
    __shared__ int      sh_cnt;
    __shared__ float    cvals[CAND];
    __shared__ int      cidx[CAND];

    int ksel = topks[n];
    if (ksel < 1) ksel = 1;
    if (ksel > 1023) ksel = 1023;

    if (threadIdx.x == 0) { sh_prefix = 0u; sh_hicount = 0; }
    __syncthreads();

    for (int pass = 3; pass >= 0; --pass) {
        for (int b = threadIdx.x; b < 256; b += 256) hist[b] = 0;
        __syncthreads();
        const unsigned prefix = sh_prefix;
        const unsigned himask = (pass == 3) ? 0u : (0xFFFFFFFFu << ((pass + 1) * 8));
        for (int v = threadIdx.x; v < VOCAB; v += 256) {
            const unsigned u = fkey(lp[v] * invT);
            if (((u ^ prefix) & himask) == 0)
                atomicAdd(&hist[(u >> (pass * 8)) & 0xFF], 1);
        }
        __syncthreads();
        if (threadIdx.x == 0) {
            const int need = ksel - sh_hicount;
            int acc = 0, bsel = 0;
            for (int b = 255; b >= 0; --b) {
                if (acc + hist[b] >= need) { bsel = b; break; }
                acc += hist[b];
            }
            sh_hicount += acc;
            sh_prefix = prefix | ((unsigned)bsel << (pass * 8));
        }
        __syncthreads();
    }
    const unsigned ustar = sh_prefix;

    if (threadIdx.x == 0) sh_cnt = 0;
    __syncthreads();
    for (int v = threadIdx.x; v < VOCAB; v += 256) {
        const float val = lp[v] * invT;
        if (fkey(val) >= ustar) {
            const int slot = atomicAdd(&sh_cnt, 1);
            if (slot < CAND) { cvals[slot] = val; cidx[slot] = v; }
        }
    }
    __syncthreads();
    int cnt = sh_cnt; if (cnt > CAND) cnt = CAND;
    for (int i = cnt + threadIdx.x; i < CAND; i += 256) {
        cvals[i] = -INFINITY; cidx[i] = 0x7FFFFFFF;
    }
    __syncthreads();

    // Bitonic sort, order = (val desc, idx asc) — matches stable argsort(-x).
    for (int k = 2; k <= CAND; k <<= 1) {
        for (int j = k >> 1; j > 0; j >>= 1) {
            for (int i = threadIdx.x; i < CAND; i += 256) {
                const int ixj = i ^ j;
                if (ixj > i) {
                    const float vi = cvals[i], vj = cvals[ixj];
                    const int   ii = cidx[i],  ij = cidx[ixj];
                    const bool iBeforeJ = (vi > vj) || (vi == vj && ii < ij);
                    const bool up = ((i & k) == 0);
                    const bool doswap = up ? !iBeforeJ : iBeforeJ;
                    if (doswap) {
                        cvals[i] = vj; cvals[ixj] = vi;
                        cidx[i]  = ij; cidx[ixj]  = ii;
                    }
                }
            }
            __syncthreads();
        }
    }

    if (threadIdx.x == 0) {
        const float p  = topps[n];
        const float mx = rowMax[n];
        const float S  = rowSum[n];
        float csum = 0.f, sk = 0.f;
        int m = 0;
        for (int j = 0; j < ksel; ++j) {
            const float e = expf(cvals[j] - mx);
            if (csum <= p) { m = j + 1; sk += e; }   // kept iff cumsum_prev <= p
            csum += e / S;
        }
        rowT[n]     = cvals[m - 1];
        rowVt[n]    = cidx[m - 1];
        rowSkept[n] = sk;
    }
}

// ---------------------------------------------------------------------------
// Kernel 5: final probs. kept iff val > t or (val == t and v <= v_t).
// ---------------------------------------------------------------------------
__global__ __launch_bounds__(256)
void sampler_out_kernel(const float* __restrict__ logits,
                        const float* __restrict__ temps,
                        const float* __restrict__ rowMax,
                        const float* __restrict__ rowT,
                        const int*   __restrict__ rowVt,
                        const float* __restrict__ rowSkept,
                        float* __restrict__ out)
{
    const int n = blockIdx.y;
    const int v = blockIdx.x * 256 + threadIdx.x;
    if (v >= VOCAB) return;
    const float val = logits[(size_t)n * VPAD + v] * (1.0f / temps[n]);
    const float t  = rowT[n];
    const int   vt = rowVt[n];
    const bool kept = (val > t) || (val == t && v <= vt);
    out[(size_t)n * VOCAB + v] = kept ? expf(val - rowMax[n]) / rowSkept[n] : 0.0f;
}

// ---------------------------------------------------------------------------
extern "C" void kernel_launch(void* const* d_in, const int* in_sizes, int n_in,
                              void* d_out, int out_size, void* d_ws, size_t ws_size,
                              hipStream_t stream)
{
    const float* hidden = (const float*)d_in[0];   // [128, 4096]
    const float* emb    = (const float*)d_in[1];   // [50257, 4096]
    const int*   toks   = (const int*)  d_in[2];   // [128, 200]
    const float* pres   = (const float*)d_in[3];   // [128]
    const float* freq   = (const float*)d_in[4];   // [128]
    const float* rep    = (const float*)d_in[5];   // [128]
    const float* temps  = (const float*)d_in[6];   // [128]
    const float* topps  = (const float*)d_in[7];   // [128]
    const int*   topks  = (const int*)  d_in[8];   // [128]
    float*       out    = (float*)d_out;           // [128, 50257]

    float* logits   = (float*)d_ws;                          // N_ROWS * VPAD
    float* rowMax   = logits + (size_t)N_ROWS * VPAD;
    float* rowSum   = rowMax + N_ROWS;
    float* rowT     = rowSum + N_ROWS;
    float* rowSkept = rowT + N_ROWS;
    int*   rowVt    = (int*)(rowSkept + N_ROWS);

    sampler_gemm_kernel<<<dim3(NB), dim3(256), 0, stream>>>(hidden, emb, logits);
    sampler_penalty_kernel<<<dim3(N_ROWS), dim3(256), 0, stream>>>(toks, pres, freq, rep, logits);
    sampler_stats_kernel<<<dim3(N_ROWS), dim3(256), 0, stream>>>(logits, temps, rowMax, rowSum);
    sampler_select_kernel<<<dim3(N_ROWS), dim3(256), 0, stream>>>(
        logits, temps, topks, topps, rowMax, rowSum, rowT, rowVt, rowSkept);
    sampler_out_kernel<<<dim3((VOCAB + 255) / 256, N_ROWS), dim3(256), 0, stream>>>(
        logits, temps, rowMax, rowT, rowVt, rowSkept, out);
}